// GraphSAGE_5497558139163
// MI455X (gfx1250) — compile-verified
//
#include <hip/hip_runtime.h>

typedef __attribute__((ext_vector_type(2))) float v2f;
typedef __attribute__((ext_vector_type(8))) float v8f;

#define FEAT 64

// ---- degree accumulation: one thread per edge ----
__global__ void deg_kernel(const int* __restrict__ dst, int nE,
                           float* __restrict__ deg) {
    int e = blockIdx.x * blockDim.x + threadIdx.x;
    if (e < nE) atomicAdd(&deg[dst[e]], 1.0f);
}

// ---- scatter-add: one thread per (edge, feature) ----
__global__ void scatter_kernel(const float* __restrict__ feat,
                               const int* __restrict__ src,
                               const int* __restrict__ dst,
                               int nE, float* __restrict__ sbuf) {
    long long t = (long long)blockIdx.x * blockDim.x + threadIdx.x;
    int e = (int)(t >> 6);
    int c = (int)(t & 63);
    if (e >= nE) return;
    int s = src[e];
    int d = dst[e];
    atomicAdd(&sbuf[(size_t)d * FEAT + c], feat[(size_t)s * FEAT + c]);
}

// ---- fused mean + dual-GEMM + bias (+ReLU) using V_WMMA_F32_16X16X4_F32 ----
// out[row, n] = (s[row]/max(deg,1)) @ Wl.T + bias + xin[row] @ Wr.T
// One wave32 handles a 16-row tile; K=64 consumed in 16 steps of K=4.
__global__ void sage_gemm_kernel(const float* __restrict__ sbuf,
                                 const float* __restrict__ deg,
                                 const float* __restrict__ xin,
                                 const float* __restrict__ Wl,
                                 const float* __restrict__ bias,
                                 const float* __restrict__ Wr,
                                 float* __restrict__ out,
                                 int n_nodes, int out_f, int do_relu) {
    const int lane = threadIdx.x & 31;
    const int wv   = threadIdx.x >> 5;
    const int wpb  = blockDim.x >> 5;
    const int tile = blockIdx.x * wpb + wv;
    const int row0 = tile * 16;
    if (row0 >= n_nodes) return;          // wave-uniform: EXEC stays all-1s inside

    const int lr    = lane & 15;          // row within tile (A) / out column (B,C/D)
    const int hi    = lane >> 4;          // half selects K pair {0,1} vs {2,3}
    const int row   = row0 + lr;
    const int kbase = hi * 2;

    const float inv = 1.0f / fmaxf(deg[row], 1.0f);

    // A-fragment layout (32-bit 16x4): lanes 0-15 hold K=k0,k0+1; lanes 16-31 hold K=k0+2,k0+3
    v2f Am[16], Ax[16];
#pragma unroll
    for (int k = 0; k < 16; ++k) {
        const float2 sm = *(const float2*)(sbuf + (size_t)row * FEAT + k * 4 + kbase);
        const float2 xm = *(const float2*)(xin  + (size_t)row * FEAT + k * 4 + kbase);
        Am[k] = v2f{sm.x * inv, sm.y * inv};
        Ax[k] = v2f{xm.x, xm.y};
    }

    const int ntiles = out_f >> 4;        // 4 for hidden(64), 2 for output(32)
    for (int t = 0; t < ntiles; ++t) {
        const int n0 = t * 16;
        v8f acc = {};
        // B-fragment: lane lr = column n0+lr; B[k][n] = W[n][k] (W row-major [out][64])
        const float* wlrow = Wl + (size_t)(n0 + lr) * FEAT + kbase;
        const float* wrrow = Wr + (size_t)(n0 + lr) * FEAT + kbase;
#pragma unroll
        for (int k = 0; k < 16; ++k) {
            const float2 w = *(const float2*)(wlrow + k * 4);
            v2f bf = {w.x, w.y};
            acc = __builtin_amdgcn_wmma_f32_16x16x4_f32(
                false, Am[k], false, bf, (short)0, acc, false, false);
        }
#pragma unroll
        for (int k = 0; k < 16; ++k) {
            const float2 w = *(const float2*)(wrrow + k * 4);
            v2f bf = {w.x, w.y};
            acc = __builtin_amdgcn_wmma_f32_16x16x4_f32(
                false, Ax[k], false, bf, (short)0, acc, false, false);
        }
        // C/D layout: VGPR r, lane = (hi ? row r+8 : row r), column n0+lr
        const float bv = bias[n0 + lr];
#pragma unroll
        for (int r = 0; r < 8; ++r) {
            float v = acc[r] + bv;
            if (do_relu) v = fmaxf(v, 0.0f);
            out[(size_t)(row0 + r + 8 * hi) * out_f + n0 + lr] = v;
        }
    }
}

extern "C" void kernel_launch(void* const* d_in, const int* in_sizes, int n_in,
                              void* d_out, int out_size, void* d_ws, size_t ws_size,
                              hipStream_t stream) {
    const float* x   = (const float*)d_in[0];
    const int*   ei1 = (const int*)d_in[1];
    const int*   ei2 = (const int*)d_in[3];
    const float* W1l = (const float*)d_in[5];
    const float* b1  = (const float*)d_in[6];
    const float* W1r = (const float*)d_in[7];
    const float* W2l = (const float*)d_in[8];
    const float* b2  = (const float*)d_in[9];
    const float* W2r = (const float*)d_in[10];

    const int N = in_sizes[0] / FEAT;     // 100000
    const int E = in_sizes[1] / 2;        // 800000 per edge list
    const int* src1 = ei1;  const int* dst1 = ei1 + E;
    const int* src2 = ei2;  const int* dst2 = ei2 + E;

    // workspace: deg[N] | s[N*64] | h[N*64]
    float* deg  = (float*)d_ws;
    float* sbuf = deg + N;
    float* hbuf = sbuf + (size_t)N * FEAT;

    hipMemsetAsync(d_ws, 0, sizeof(float) * (size_t)N * (1 + FEAT), stream);

    const int TB = 256;
    const int deg_blocks = (E + TB - 1) / TB;
    deg_kernel<<<deg_blocks, TB, 0, stream>>>(dst1, E, deg);
    deg_kernel<<<deg_blocks, TB, 0, stream>>>(dst2, E, deg);

    const long long sc_threads = (long long)E * FEAT;
    const int sc_blocks = (int)((sc_threads + TB - 1) / TB);
    scatter_kernel<<<sc_blocks, TB, 0, stream>>>(x, src1, dst1, E, sbuf);
    scatter_kernel<<<sc_blocks, TB, 0, stream>>>(x, src2, dst2, E, sbuf);

    // layer 1: h = relu(mean @ W1l.T + b1 + x @ W1r.T)
    const int waves  = (N + 15) / 16;     // 6250 full 16-row tiles
    const int wpb    = 8;                 // 256 threads = 8 waves per block
    const int gblks  = (waves + wpb - 1) / wpb;
    sage_gemm_kernel<<<gblks, wpb * 32, 0, stream>>>(
        sbuf, deg, x, W1l, b1, W1r, hbuf, N, FEAT, 1);

    // layer 2 aggregation on h
    hipMemsetAsync(sbuf, 0, sizeof(float) * (size_t)N * FEAT, stream);
    scatter_kernel<<<sc_blocks, TB, 0, stream>>>(hbuf, src1, dst1, E, sbuf);
    scatter_kernel<<<sc_blocks, TB, 0, stream>>>(hbuf, src2, dst2, E, sbuf);

    // layer 2: out = mean2 @ W2l.T + b2 + h @ W2r.T  (no relu), out_f = 32
    sage_gemm_kernel<<<gblks, wpb * 32, 0, stream>>>(
        sbuf, deg, hbuf, W2l, b2, W2r, (float*)d_out, N, 32, 0);
}